// SEBlock_23235773072056
// MI455X (gfx1250) — compile-verified
//
#include <hip/hip_runtime.h>
#include <math.h>

// ---------------------------------------------------------------------------
// SE block: segment-mean -> tiny MLP (WMMA f32 16x16x4) -> sigmoid gate -> mul
// MI455X reasoning: 1.5 GB of HBM traffic @ 23.3 TB/s => ~66us floor; the MLP
// is ~50 KFLOPs so it runs in one wave32 using V_WMMA_F32_16X16X4_F32.
// ---------------------------------------------------------------------------

#define NB  16   // NUM_BATCHES
#define CH  64   // channels
#define HID 16   // hidden

typedef __attribute__((ext_vector_type(2))) float v2f;
typedef __attribute__((ext_vector_type(8))) float v8f;

// ws layout (floats): [0,1024) seg_sum[16][64], [1024,1040) counts[16],
//                     [1040,2064) gate[16][64]

__global__ void se_zero(float* __restrict__ p, int n) {
    int i = blockIdx.x * blockDim.x + threadIdx.x;
    if (i < n) p[i] = 0.0f;
}

// ---------------------------------------------------------------------------
// Pass 1: segment sum. batch_idx is sorted, so each thread keeps a running
// (batch, acc) in registers and only flushes to LDS on a batch transition;
// one global float atomic per (batch,channel) per block at the end.
// Thread t handles channel (t&63); 4 point-streams per 256-thread block.
// ---------------------------------------------------------------------------
__global__ void se_segsum(const float* __restrict__ feats,
                          const int* __restrict__ bidx,
                          float* __restrict__ seg_sum,
                          float* __restrict__ cnt, int N) {
    __shared__ float s_sum[NB * CH];
    __shared__ float s_cnt[NB];
    const int t = threadIdx.x;
    for (int i = t; i < NB * CH; i += blockDim.x) s_sum[i] = 0.0f;
    if (t < NB) s_cnt[t] = 0.0f;
    __syncthreads();

    const int c          = t & (CH - 1);
    const int prow       = t >> 6;                    // 0..3
    const int rows_blk   = blockDim.x >> 6;           // 4
    long p               = (long)blockIdx.x * rows_blk + prow;
    const long pstride   = (long)gridDim.x * rows_blk;

    float acc = 0.0f, cacc = 0.0f;
    int   cur = -1;
    for (; p < N; p += pstride) {
        int b = bidx[p];
        if (b != cur) {                               // rare: batch_idx sorted
            if (cur >= 0) {
                atomicAdd(&s_sum[cur * CH + c], acc);
                if (c == 0) atomicAdd(&s_cnt[cur], cacc);
            }
            acc = 0.0f; cacc = 0.0f; cur = b;
        }
        acc  += feats[p * CH + c];
        cacc += 1.0f;
    }
    if (cur >= 0) {
        atomicAdd(&s_sum[cur * CH + c], acc);
        if (c == 0) atomicAdd(&s_cnt[cur], cacc);
    }
    __syncthreads();
    for (int i = t; i < NB * CH; i += blockDim.x) atomicAdd(&seg_sum[i], s_sum[i]);
    if (t < NB) atomicAdd(&cnt[t], s_cnt[t]);
}

// ---------------------------------------------------------------------------
// Pass 2: single wave32 does mean + MLP + sigmoid via V_WMMA_F32_16X16X4_F32.
//   GEMM1: H[16x16]  = mean[16x64] @ W1[64x16]   -> 16 WMMA (K=64 in 4-chunks)
//   GEMM2: G[16x64]  = H[16x16]    @ W2[16x64]   -> 4 N-tiles x 4 WMMA
// A layout (ISA 7.12.2, 32-bit A 16x4): lanes 0-15 = M, {a.x,a.y} = K{0,1};
// lanes 16-31 = M, K{2,3}. B mirrored with lane = N. C/D: VGPR r holds
// M = r + 8*(lane>=16), N = lane&15.
// ---------------------------------------------------------------------------
__global__ void __launch_bounds__(32)
se_mlp_wmma(const float* __restrict__ seg_sum, const float* __restrict__ cnt,
            const float* __restrict__ W1, const float* __restrict__ b1,
            const float* __restrict__ W2, const float* __restrict__ b2,
            float* __restrict__ gate) {
    __shared__ float s_mean[NB * CH];
    __shared__ float s_h[NB * HID];
    const int lane = threadIdx.x;
    const int half = lane >> 4;     // 0 or 1
    const int l16  = lane & 15;

    // mean = seg_sum / max(count,1)
    for (int i = lane; i < NB * CH; i += 32) {
        float n = cnt[i / CH];
        s_mean[i] = seg_sum[i] / fmaxf(n, 1.0f);
    }
    __syncthreads();

    // --- GEMM1: K = 64 as 16 steps of K=4 ---
    v8f c1 = {};
    for (int kb = 0; kb < 16; ++kb) {
        const int k0 = kb * 4 + half * 2;
        v2f a, b;
        a.x = s_mean[l16 * CH + k0];
        a.y = s_mean[l16 * CH + k0 + 1];
        b.x = W1[(k0    ) * HID + l16];
        b.y = W1[(k0 + 1) * HID + l16];
        c1 = __builtin_amdgcn_wmma_f32_16x16x4_f32(false, a, false, b,
                                                   (short)0, c1, false, false);
    }
    // bias + relu; stage H in LDS in row-major for A-layout reload
    const float bias1 = b1[l16];
    #pragma unroll
    for (int r = 0; r < 8; ++r) {
        float h = fmaxf(c1[r] + bias1, 0.0f);
        s_h[(r + 8 * half) * HID + l16] = h;
    }
    __syncthreads();

    // --- GEMM2: four 16-wide N tiles, K = 16 as 4 steps of K=4 ---
    for (int nt = 0; nt < 4; ++nt) {
        v8f c2 = {};
        #pragma unroll
        for (int kb = 0; kb < 4; ++kb) {
            const int k0 = kb * 4 + half * 2;
            v2f a, b;
            a.x = s_h[l16 * HID + k0];
            a.y = s_h[l16 * HID + k0 + 1];
            b.x = W2[(k0    ) * CH + nt * 16 + l16];
            b.y = W2[(k0 + 1) * CH + nt * 16 + l16];
            c2 = __builtin_amdgcn_wmma_f32_16x16x4_f32(false, a, false, b,
                                                       (short)0, c2, false, false);
        }
        const float bias2 = b2[nt * 16 + l16];
        #pragma unroll
        for (int r = 0; r < 8; ++r) {
            float g = c2[r] + bias2;
            g = 1.0f / (1.0f + __expf(-g));            // sigmoid
            gate[(r + 8 * half) * CH + nt * 16 + l16] = g;
        }
    }
}

// ---------------------------------------------------------------------------
// Pass 3: out = feats * gate[batch_idx].  float4-vectorized stream; the 4 KB
// gate table lives in LDS. 16 float4 per point.
// ---------------------------------------------------------------------------
__global__ void se_apply(const float* __restrict__ feats,
                         const int* __restrict__ bidx,
                         const float* __restrict__ gate,
                         float* __restrict__ out, long n4) {
    __shared__ float s_gate[NB * CH];
    for (int i = threadIdx.x; i < NB * CH; i += blockDim.x) s_gate[i] = gate[i];
    __syncthreads();

    long i            = (long)blockIdx.x * blockDim.x + threadIdx.x;
    const long stride = (long)gridDim.x * blockDim.x;
    const float4* f4  = (const float4*)feats;
    float4* o4        = (float4*)out;

    for (; i < n4; i += stride) {
        const long pt = i >> 4;                 // point index
        const int  c4 = (int)(i & 15);          // float4 slot within point
        __builtin_prefetch(&f4[i + stride], 0, 0);   // global_prefetch_b8
        const int  b  = bidx[pt];
        float4 v = f4[i];
        const float* g = &s_gate[b * CH + c4 * 4];
        v.x *= g[0]; v.y *= g[1]; v.z *= g[2]; v.w *= g[3];
        o4[i] = v;
    }
}

// ---------------------------------------------------------------------------
extern "C" void kernel_launch(void* const* d_in, const int* in_sizes, int n_in,
                              void* d_out, int out_size, void* d_ws, size_t ws_size,
                              hipStream_t stream) {
    const float* feats = (const float*)d_in[0];
    const int*   bidx  = (const int*)d_in[1];
    const float* W1    = (const float*)d_in[2];
    const float* b1    = (const float*)d_in[3];
    const float* W2    = (const float*)d_in[4];
    const float* b2    = (const float*)d_in[5];
    float*       out   = (float*)d_out;
    const int    N     = in_sizes[1];           // batch_idx has N entries

    float* ws      = (float*)d_ws;
    float* seg_sum = ws;                        // 1024 floats
    float* cnt     = ws + NB * CH;              // 16 floats
    float* gate    = ws + NB * CH + NB;         // 1024 floats

    // Zero accumulators every call (deterministic across graph replays).
    se_zero<<<(NB * CH + NB + 255) / 256, 256, 0, stream>>>(ws, NB * CH + NB);

    se_segsum<<<512, 256, 0, stream>>>(feats, bidx, seg_sum, cnt, N);

    se_mlp_wmma<<<1, 32, 0, stream>>>(seg_sum, cnt, W1, b1, W2, b2, gate);

    const long n4 = (long)N * (CH / 4);
    long want = (n4 + 255) / 256;
    int blocks = (int)(want > 16384 ? 16384 : want);
    se_apply<<<blocks, 256, 0, stream>>>(feats, bidx, gate, out, n4);
}